// DynamicRangeCompressor_83090437308532
// MI455X (gfx1250) — compile-verified
//
#include <hip/hip_runtime.h>
#include <hip/hip_bf16.h>
#include <math.h>

typedef __attribute__((ext_vector_type(2))) float v2f;
typedef __attribute__((ext_vector_type(8))) float v8f;

#define BATCH   16
#define TLEN    2097152
#define FRAMES  131072        // TLEN / 16
#define CHUNK   2048
#define NCHUNK  (FRAMES / CHUNK)   // 64 chunks per batch
#define WARM    2048               // warm-up length; error ~ a^2048 -> below fp32 eps

// ---------------------------------------------------------------------------
// Kernel 1: per-frame static gain (in dB) + 16x linear downsample.
// src = 16f + 7.5 exactly, so gd[f] = 0.5*(gain[16f+7] + gain[16f+8]).
// gain(x) = -max(20*log10(|x|+1e-8) - thr, 0) * (1 - 1/ratio)
// ---------------------------------------------------------------------------
__global__ __launch_bounds__(256) void drc_gain_ds(const float* __restrict__ audio,
                                                   const float* __restrict__ thr_p,
                                                   const float* __restrict__ ratio_p,
                                                   float* __restrict__ gd)
{
    int idx = blockIdx.x * 256 + threadIdx.x;        // exactly BATCH*FRAMES threads
    int b = idx >> 17;                               // / FRAMES
    int f = idx & (FRAMES - 1);
    float thr = thr_p[0];
    float k   = 1.0f - 1.0f / ratio_p[0];
    const float* ab = audio + (size_t)b * TLEN;
    int t0 = (f << 4) + 7;
    float x0 = ab[t0];
    float x1 = ab[t0 + 1];
    const float C = 20.0f * 0.30102999566f;          // log10 via log2
    float db0 = C * __log2f(fabsf(x0) + 1e-8f);
    float db1 = C * __log2f(fabsf(x1) + 1e-8f);
    float g0 = -fmaxf(db0 - thr, 0.0f) * k;
    float g1 = -fmaxf(db1 - thr, 0.0f) * k;
    gd[idx] = 0.5f * (g0 + g1);
}

// ---------------------------------------------------------------------------
// Kernel 2: attack/release one-pole smoothing. Chunked parallel scan with
// exponential-forgetting warm-up (chunk 0 exact; others warm-start WARM frames
// early). new = a*prev + (1-a)*t  ==  fma(a, prev - t, t).
// 16 batches * 64 chunks = 1024 independent serial threads of <= 4096 steps.
// ---------------------------------------------------------------------------
__global__ __launch_bounds__(256) void drc_scan(const float* __restrict__ gd,
                                                float* __restrict__ gs,
                                                const float* __restrict__ att_p,
                                                const float* __restrict__ rel_p)
{
    int tid = blockIdx.x * 256 + threadIdx.x;        // 0 .. BATCH*NCHUNK-1
    int b = tid / NCHUNK;
    int c = tid - b * NCHUNK;
    float att = att_p[0];
    float rel = rel_p[0];
    const float* g = gd + b * FRAMES;
    float*       o = gs + b * FRAMES;
    int start = c * CHUNK;
    int f0 = (c == 0) ? 0 : (start - WARM);
    float prev = g[f0];                              // scan init: ys[0] == gd[0]
#pragma unroll 4
    for (int f = f0; f < start; ++f) {               // warm-up (no writes)
        float t = g[f];
        float a = (t >= prev) ? att : rel;
        prev = __builtin_fmaf(a, prev - t, t);
    }
#pragma unroll 4
    for (int f = start; f < start + CHUNK; ++f) {
        float t = g[f];
        float a = (t >= prev) ? att : rel;
        prev = __builtin_fmaf(a, prev - t, t);
        o[f] = prev;
    }
}

// ---------------------------------------------------------------------------
// Kernel 3: Hann overlap-add upsample as a WMMA f32 16x16x4 matmul + output.
// gup[16q + r] = gs[q]*w1[r] + gs[q+1]*w2[r],  w1[r] = 0.5*(1+cos(pi*r/16)).
// Per wave: A(16x4) rows = [gs[fb+M], gs[fb+M+1], 0, 0]  (K0/K1 live in lanes
// 0-15 of the two A VGPRs; K2/K3 lanes 16-31 zero), B(4x16) rows = [w1; w2; 0; 0].
// D(16x16) = one 16-frame x 16-sample tile of gup.
// out = sign(audio) * (|audio|+1e-8) * 10^((gup+makeup)/20).
// ---------------------------------------------------------------------------
__global__ __launch_bounds__(256) void drc_upsample_out(const float* __restrict__ audio,
                                                        const float* __restrict__ gs,
                                                        const float* __restrict__ makeup_p,
                                                        float* __restrict__ out)
{
    float makeup = makeup_p[0];
    int lane = threadIdx.x & 31;
    int wave = blockIdx.x * 8 + (threadIdx.x >> 5); // 8 waves / block, no early exit
    int b    = wave >> 13;                           // 8192 tiles per batch
    int tile = wave & 8191;
    int fb   = tile << 4;                            // first frame of tile

    const float* gsb = gs + b * FRAMES;

    // A matrix (16x4 f32): v0 = K0 (lanes 0-15) / K2 (lanes 16-31, zero)
    //                      v1 = K1 (lanes 0-15) / K3 (lanes 16-31, zero)
    float a0 = 0.0f, a1 = 0.0f;
    // B matrix (4x16 f32): v0 = row K0 (lanes 0-15) / row K2 (zero)
    //                      v1 = row K1 (lanes 0-15) / row K3 (zero)
    float b0 = 0.0f, b1 = 0.0f;
    if (lane < 16) {
        int q  = fb + lane;
        int q1 = q + 1;  if (q1 > FRAMES - 1) q1 = FRAMES - 1;   // endpoint clamp
        a0 = gsb[q];
        a1 = gsb[q1];
        float w1 = 0.5f * (1.0f + __cosf((float)lane * (3.14159265358979f / 16.0f)));
        b0 = w1;
        b1 = 1.0f - w1;
    }
    v2f A;  A.x = a0;  A.y = a1;
    v2f Bm; Bm.x = b0; Bm.y = b1;
    v8f Cz = {};
    // D = A x B + 0   -> v_wmma_f32_16x16x4_f32
    v8f D = __builtin_amdgcn_wmma_f32_16x16x4_f32(
        /*neg_a=*/false, A, /*neg_b=*/false, Bm,
        /*c_mod=*/(short)0, Cz, /*reuse_a=*/false, /*reuse_b=*/false);

    // D layout: VGPR v, lanes 0-15 -> M=v, N=lane; lanes 16-31 -> M=v+8, N=lane-16
    int mbase = fb + ((lane >> 4) << 3);
    int r     = lane & 15;
    const float* ab = audio + (size_t)b * TLEN;
    float*       ob = out   + (size_t)b * TLEN;
    const float E = 2.30258509299f / 20.0f;          // ln(10)/20: 10^(y/20)=e^(y*E)
#pragma unroll
    for (int v = 0; v < 8; ++v) {
        int t = ((mbase + v) << 4) + r;
        float x   = ab[t];
        float gup = D[v];
        float mag = (fabsf(x) + 1e-8f) * __expf((gup + makeup) * E);
        ob[t] = (x < 0.0f) ? -mag : mag;
    }
}

extern "C" void kernel_launch(void* const* d_in, const int* in_sizes, int n_in,
                              void* d_out, int out_size, void* d_ws, size_t ws_size,
                              hipStream_t stream)
{
    const float* audio  = (const float*)d_in[0];
    const float* thr    = (const float*)d_in[1];
    const float* ratio  = (const float*)d_in[2];
    const float* makeup = (const float*)d_in[3];
    const float* att    = (const float*)d_in[4];
    const float* rel    = (const float*)d_in[5];
    float* out = (float*)d_out;

    // workspace: gd (8 MB) then gs (8 MB)
    float* gd = (float*)d_ws;
    float* gs = gd + (size_t)BATCH * FRAMES;

    // 1) gain + 16x downsample: one thread per (b, frame)
    drc_gain_ds<<<(BATCH * FRAMES) / 256, 256, 0, stream>>>(audio, thr, ratio, gd);

    // 2) chunked attack/release scan: 1024 serial threads
    drc_scan<<<(BATCH * NCHUNK) / 256, 256, 0, stream>>>(gd, gs, att, rel);

    // 3) WMMA overlap-add upsample + output: one wave per 16-frame tile
    drc_upsample_out<<<(BATCH * (FRAMES / 16)) / 8, 256, 0, stream>>>(audio, gs, makeup, out);
}